// GCN_audio_fea_37434934952162
// MI455X (gfx1250) — compile-verified
//
#include <hip/hip_runtime.h>

// ---------------------------------------------------------------------------
// GCN audio feature net for MI455X (gfx1250), compile-only tuning loop.
//
// Roofline: dominant cost is streaming sounds (256 MB) for the channel sum
// (~11 us @ 23.3 TB/s) plus ~0.2 GB scatter-gather for the 16 node columns.
// Total FLOPs (~2.7 GFLOP) are tiny -> keep full fp32 and use
// V_WMMA_F32_16X16X4_F32 for both GEMMs (exact fp32, tensor-pipe).
// Node gather overlaps WMMA compute via CDNA5 GLOBAL_LOAD_ASYNC_TO_LDS_B32
// with a double-buffered LDS tile (ASYNCcnt pipeline).
// ---------------------------------------------------------------------------

#define C_IN    2048
#define C_OUT   256
#define KNODES  16
#define NBATCH  128
#define WH      256     // 16*16 spatial sites
#define D_OUT   512
#define KFC     4096    // KNODES * C_OUT

#define CHUNK_K     512
#define NCHUNKS     (C_IN / CHUNK_K)
#define NODE_STRIDE 17   // pad k-rows of nodes tile (16 -> 17 floats): conflict-free
#define X_STRIDE    260  // pad x rows (256 -> 260 floats)

typedef __attribute__((ext_vector_type(2))) float v2f;
typedef __attribute__((ext_vector_type(8))) float v8f;

#if defined(__gfx1250__) &&                                              \
    __has_builtin(__builtin_amdgcn_global_load_async_to_lds_b32) &&      \
    __has_builtin(__builtin_amdgcn_s_wait_asynccnt)
#define USE_ASYNC 1
typedef __attribute__((address_space(1))) int async_gptr_t;
typedef __attribute__((address_space(3))) int async_lptr_t;
#else
#define USE_ASYNC 0
#endif

// ---------------------------------------------------------------------------
// Kernel 1: fre[n, s] = sum_c sounds[n, c, s]   (memory-bound streaming)
// One block per image; 256 threads = 256 spatial sites; coalesced 1 KB rows.
// ---------------------------------------------------------------------------
__global__ __launch_bounds__(256) void chansum_kernel(
    const float* __restrict__ sounds, float* __restrict__ sums) {
  const int n = blockIdx.x;
  const int s = threadIdx.x;
  const float* p = sounds + (size_t)n * C_IN * WH + s;
  float acc = 0.f;
#pragma unroll 8
  for (int c = 0; c < C_IN; ++c) acc += p[(size_t)c * WH];
  sums[n * WH + s] = acc;
}

// ---------------------------------------------------------------------------
// Kernel 2: per image: top-16 (jax.lax.top_k order: descending value, lowest
// index wins ties), row/col remap, spatial offsets, and symmetric-normalized
// graph lnorm = D^{-1/2} A D^{-1/2}, A = exp(-d2/32).
// One wave (32 threads) per image.
// ---------------------------------------------------------------------------
__global__ __launch_bounds__(32) void topk_lnorm_kernel(
    const float* __restrict__ sums, int* __restrict__ s_idx,
    float* __restrict__ lnorm) {
  const int n = blockIdx.x;
  const int t = threadIdx.x;
  __shared__ float v[WH];
  __shared__ int   sel[KNODES];
  __shared__ float cr[KNODES], cc[KNODES];
  __shared__ float Ash[KNODES * KNODES];
  __shared__ float dinv[KNODES];

  for (int i = t; i < WH; i += 32) v[i] = sums[n * WH + i];
  __syncthreads();

  for (int k = 0; k < KNODES; ++k) {
    float bv = -3.402823466e38f;
    int   bi = WH;
    // each lane scans 8 contiguous entries; strict '>' keeps lowest index
#pragma unroll
    for (int i = 0; i < 8; ++i) {
      int idx = t * 8 + i;
      float x = v[idx];
      if (x > bv) { bv = x; bi = idx; }
    }
    // wave32 reduction: prefer larger value, then smaller index
    for (int off = 16; off > 0; off >>= 1) {
      float ov = __shfl_down(bv, off);
      int   oi = __shfl_down(bi, off);
      if (ov > bv || (ov == bv && oi < bi)) { bv = ov; bi = oi; }
    }
    bi = __shfl(bi, 0);
    if (t == 0) { sel[k] = bi; v[bi] = -3.402823466e38f; }
    __syncthreads();
  }

  if (t < KNODES) {
    int idx = sel[t];
    int r = (idx < 16) ? 0 : (idx / 16 - 1);
    int cm = idx & 15;
    int c = (cm == 0) ? 15 : (cm - 1);
    s_idx[n * KNODES + t] = r * 16 + c;
    cr[t] = (float)r;
    cc[t] = (float)c;
  }
  __syncthreads();
  for (int e = t; e < KNODES * KNODES; e += 32) {
    int i = e >> 4, j = e & 15;
    float dr = cr[i] - cr[j], dc = cc[i] - cc[j];
    Ash[e] = expf(-(dr * dr + dc * dc) * (1.0f / 32.0f));
  }
  __syncthreads();
  if (t < KNODES) {
    float sum = 0.f;
#pragma unroll
    for (int j = 0; j < KNODES; ++j) sum += Ash[t * KNODES + j];
    dinv[t] = 1.0f / sqrtf(sum);
  }
  __syncthreads();
  for (int e = t; e < KNODES * KNODES; e += 32) {
    int i = e >> 4, j = e & 15;
    lnorm[n * 256 + e] = Ash[e] * dinv[i] * dinv[j];
  }
}

// ---------------------------------------------------------------------------
// Stage one 512-k chunk of the 16 gathered node columns into LDS.
// Thread owns node m = tid&15 and k-rows kk = (tid>>4) + i*16.
// Async path: GLOBAL_LOAD_ASYNC_TO_LDS_B32 (no VGPR round trip, ASYNCcnt).
// ---------------------------------------------------------------------------
__device__ __forceinline__ void stage_nodes(const float* __restrict__ gm,
                                            float* __restrict__ buf,
                                            int kc, int kkbase, int m) {
#if USE_ASYNC
#pragma unroll
  for (int i = 0; i < (CHUNK_K * KNODES) / 256; ++i) {
    const int kk = kkbase + i * 16;
    const float* g = gm + (size_t)(kc + kk) * WH;
    float* l = &buf[kk * NODE_STRIDE + m];
    __builtin_amdgcn_global_load_async_to_lds_b32(
        (async_gptr_t*)g, (async_lptr_t*)l, 0, 0);
  }
#else
#pragma unroll 8
  for (int i = 0; i < (CHUNK_K * KNODES) / 256; ++i) {
    const int kk = kkbase + i * 16;
    buf[kk * NODE_STRIDE + m] = gm[(size_t)(kc + kk) * WH];
  }
#endif
}

__device__ __forceinline__ void wait_stage_done() {
#if USE_ASYNC
  __builtin_amdgcn_s_wait_asynccnt(0);
#endif
  __syncthreads();
}

// ---------------------------------------------------------------------------
// Kernel 3: per image n:
//   nodes[16,2048] gathered from sounds via s_idx; x = relu(nodes @ conv_w^T
//   + b) with V_WMMA_F32_16X16X4_F32 (M=16 nodes, N=256 out-ch, K=2048);
//   y = lnorm @ x (tiny, VALU); write Y[n, k*256+o].
// 256 threads = 8 waves; wave w owns out-channel tiles [32w,32w+16) and [32w+16,32w+32).
// nodes double-buffered in LDS per 512-k chunk ([k][m] layout, row stride 17);
// chunk c+1's async gather overlaps chunk c's WMMA loop.
// WMMA fragment layouts (CDNA5 ISA 7.12.2):
//   A 16x4  f32: vgpr=K%2, lane = M + 16*(K/2)
//   B 4x16  f32: vgpr=K%2, lane = N + 16*(K/2)
//   C 16x16 f32: vgpr=M%8, lane = N + 16*(M/8)
// ---------------------------------------------------------------------------
__global__ __launch_bounds__(256) void conv_graph_kernel(
    const float* __restrict__ sounds, const float* __restrict__ conv_w,
    const float* __restrict__ conv_b, const int* __restrict__ s_idx,
    const float* __restrict__ lnorm, float* __restrict__ Y) {
  __shared__ float nodes_lds[2][CHUNK_K * NODE_STRIDE];
  __shared__ float x_lds[KNODES * X_STRIDE];
  __shared__ float ln_lds[256];
  __shared__ int   s_lds[KNODES];

  const int n    = blockIdx.x;
  const int tid  = threadIdx.x;
  const int lane = tid & 31;
  const int wave = tid >> 5;
  const int half = lane >> 4;  // which 16-lane half
  const int lm   = lane & 15;

  if (tid < KNODES) s_lds[tid] = s_idx[n * KNODES + tid];
  ln_lds[tid] = lnorm[n * 256 + tid];
  __syncthreads();

  const float* sbase  = sounds + (size_t)n * C_IN * WH;
  const int    m      = tid & 15;
  const int    kkbase = tid >> 4;
  const float* gm     = sbase + s_lds[m];

  const int oA = wave * 32;
  const int oB = oA + 16;

  v8f acc0 = {}; v8f acc1 = {};

  // prologue: kick off chunk 0
  stage_nodes(gm, nodes_lds[0], 0, kkbase, m);

  for (int c = 0; c < NCHUNKS; ++c) {
    wait_stage_done();  // s_wait_asynccnt 0 + barrier: buf[c&1] complete,
                        // and all waves done reading buf[(c+1)&1]
    if (c + 1 < NCHUNKS)
      stage_nodes(gm, nodes_lds[(c + 1) & 1], (c + 1) * CHUNK_K, kkbase, m);

    const float* buf = nodes_lds[c & 1];
    const int kc = c * CHUNK_K;
#pragma unroll 4
    for (int k0 = 0; k0 < CHUNK_K; k0 += 4) {
      const int kr = k0 + 2 * half;
      v2f a;
      a.x = buf[kr * NODE_STRIDE + lm];
      a.y = buf[(kr + 1) * NODE_STRIDE + lm];
      const int kg = kc + kr;
      v2f b0 = *(const v2f*)&conv_w[(size_t)(oA + lm) * C_IN + kg];
      v2f b1 = *(const v2f*)&conv_w[(size_t)(oB + lm) * C_IN + kg];
      acc0 = __builtin_amdgcn_wmma_f32_16x16x4_f32(false, a, false, b0,
                                                   (short)0, acc0, false, false);
      acc1 = __builtin_amdgcn_wmma_f32_16x16x4_f32(false, a, false, b1,
                                                   (short)0, acc1, false, false);
    }
  }

  // epilogue: x = relu(acc + bias) into LDS (C layout -> [m][o])
  const float bA = conv_b[oA + lm];
  const float bB = conv_b[oB + lm];
#pragma unroll
  for (int v = 0; v < 8; ++v) {
    int mm = v + 8 * half;
    float xa = acc0[v] + bA;
    float xb = acc1[v] + bB;
    x_lds[mm * X_STRIDE + oA + lm] = xa > 0.f ? xa : 0.f;
    x_lds[mm * X_STRIDE + oB + lm] = xb > 0.f ? xb : 0.f;
  }
  __syncthreads();

  // y[i][o] = sum_j lnorm[i][j] * x[j][o];  o = tid (coalesced LDS + store)
  float* yout = Y + (size_t)n * KFC;
#pragma unroll
  for (int i = 0; i < KNODES; ++i) {
    float acc = 0.f;
#pragma unroll
    for (int j = 0; j < KNODES; ++j)
      acc += ln_lds[i * KNODES + j] * x_lds[j * X_STRIDE + tid];
    yout[i * C_OUT + tid] = acc;
  }
}

// ---------------------------------------------------------------------------
// Kernel 4: out[128,512] = Y[128,4096] @ fc_w^T + fc_b  via fp32 WMMA.
// 8 blocks = M-tiles of 16 rows; 8 waves x 4 N-tiles of 16; K=4096.
// A and B fragments are per-lane sequential float2 streams (L1/L2 friendly;
// fc_w is 8 MB and stays resident in the 192 MB L2).
// ---------------------------------------------------------------------------
__global__ __launch_bounds__(256) void fc_kernel(
    const float* __restrict__ Y, const float* __restrict__ fc_w,
    const float* __restrict__ fc_b, float* __restrict__ out) {
  const int tid  = threadIdx.x;
  const int lane = tid & 31;
  const int wave = tid >> 5;
  const int half = lane >> 4;
  const int lm   = lane & 15;
  const int nrow = blockIdx.x * 16;

  v8f acc[4] = {{}, {}, {}, {}};
  const float* arow = Y + (size_t)(nrow + lm) * KFC;

#pragma unroll 2
  for (int k0 = 0; k0 < KFC; k0 += 4) {
    const int kg = k0 + 2 * half;
    v2f a = *(const v2f*)&arow[kg];
#pragma unroll
    for (int t = 0; t < 4; ++t) {
      const int d = wave * 64 + t * 16 + lm;
      v2f b = *(const v2f*)&fc_w[(size_t)d * KFC + kg];
      acc[t] = __builtin_amdgcn_wmma_f32_16x16x4_f32(false, a, false, b,
                                                     (short)0, acc[t], false, false);
    }
  }

#pragma unroll
  for (int t = 0; t < 4; ++t) {
    const int d = wave * 64 + t * 16 + lm;
    const float bias = fc_b[d];
#pragma unroll
    for (int v = 0; v < 8; ++v) {
      int mrow = v + 8 * half;
      out[(size_t)(nrow + mrow) * D_OUT + d] = acc[t][v] + bias;
    }
  }
}

// ---------------------------------------------------------------------------
// Launch. Workspace layout (bytes):
//   [0,        131072) sums   128*256 f32
//   [131072,   139264) s_idx  128*16  i32
//   [139264,   270336) lnorm  128*256 f32
//   [270336,  2367488) Y      128*4096 f32
// ---------------------------------------------------------------------------
extern "C" void kernel_launch(void* const* d_in, const int* in_sizes, int n_in,
                              void* d_out, int out_size, void* d_ws, size_t ws_size,
                              hipStream_t stream) {
  const float* sounds = (const float*)d_in[0];
  const float* conv_w = (const float*)d_in[1];
  const float* conv_b = (const float*)d_in[2];
  const float* fc_w   = (const float*)d_in[3];
  const float* fc_b   = (const float*)d_in[4];
  float* out = (float*)d_out;

  char* ws = (char*)d_ws;
  float* sums  = (float*)(ws);
  int*   sidx  = (int*)(ws + 131072);
  float* lnorm = (float*)(ws + 139264);
  float* Yb    = (float*)(ws + 270336);

  chansum_kernel<<<NBATCH, 256, 0, stream>>>(sounds, sums);
  topk_lnorm_kernel<<<NBATCH, 32, 0, stream>>>(sums, sidx, lnorm);
  conv_graph_kernel<<<NBATCH, 256, 0, stream>>>(sounds, conv_w, conv_b, sidx,
                                                lnorm, Yb);
  fc_kernel<<<NBATCH / 16, 256, 0, stream>>>(Yb, fc_w, fc_b, out);
}